// RoiAlign_52235392254190
// MI455X (gfx1250) — compile-verified
//
#include <hip/hip_runtime.h>
#include <cstdint>

// RoiAlign (multi-level crop_and_resize, bilinear, extrapolation_value=0)
// B=2, N=512, C=256, POOL=7, levels {256,128,64,32}, float32 in/out.
//
// Memory-bound gather kernel (AI ~0.55 FLOP/B): no matmul structure -> no WMMA.
// gfx1250 paths used instead: global_load_async_to_lds_b128 double-buffered
// pipeline (ASYNCcnt), s_wait_asynccnt, non-temporal b128 stores.

typedef float v4f __attribute__((ext_vector_type(4)));

#define POOL   7
#define NPTS   49      // POOL*POOL
#define NBOX   512
#define NLVL   4

struct Pt {
  unsigned long long a[4];   // byte addresses of the 4 neighbor pixel rows (+chan offset)
  float w00, w01, w10, w11;  // bilinear weights
  int   valid;               // sample inside [0, S-1] on both axes
};

__device__ __forceinline__ Pt make_pt(unsigned long long fbase, int S, int b,
                                      float ybase, float dy, float xbase, float dx,
                                      int p, int c4) {
  Pt r;
  const int i = p / POOL;
  const int j = p - i * POOL;
  const float ys = fmaf((float)i, dy, ybase);
  const float xs = fmaf((float)j, dx, xbase);
  const float yf = floorf(ys), xf = floorf(xs);
  const float fy = ys - yf,   fx = xs - xf;
  int y0 = (int)yf; y0 = min(max(y0, 0), S - 1);
  const int y1 = min(y0 + 1, S - 1);
  int x0 = (int)xf; x0 = min(max(x0, 0), S - 1);
  const int x1 = min(x0 + 1, S - 1);
  const float sm1 = (float)(S - 1);
  r.valid = (ys >= 0.f) && (ys <= sm1) && (xs >= 0.f) && (xs <= sm1);

  const unsigned long long cb = ((unsigned long long)c4) << 4;  // 16B chunk per thread
  const unsigned long long r0 = (unsigned long long)(b * S + y0) * (unsigned long long)S;
  const unsigned long long r1 = (unsigned long long)(b * S + y1) * (unsigned long long)S;
  r.a[0] = fbase + (((r0 + (unsigned long long)x0) << 10)) + cb;  // <<10: 256 ch * 4B
  r.a[1] = fbase + (((r0 + (unsigned long long)x1) << 10)) + cb;
  r.a[2] = fbase + (((r1 + (unsigned long long)x0) << 10)) + cb;
  r.a[3] = fbase + (((r1 + (unsigned long long)x1) << 10)) + cb;

  const float gy = 1.f - fy, gx = 1.f - fx;
  r.w00 = gy * gx; r.w01 = gy * fx; r.w10 = fy * gx; r.w11 = fy * fx;
  return r;
}

// Async-gather this thread's 16B chunk of each of the 4 neighbor pixel rows
// into LDS (rows spaced 4096B apart in the staging buffer).
__device__ __forceinline__ void issue4(const Pt& pt, unsigned ldsb) {
  asm volatile("global_load_async_to_lds_b128 %0, %1, off"
               :: "v"(ldsb),           "v"(pt.a[0]) : "memory");
  asm volatile("global_load_async_to_lds_b128 %0, %1, off"
               :: "v"(ldsb + 4096u),   "v"(pt.a[1]) : "memory");
  asm volatile("global_load_async_to_lds_b128 %0, %1, off"
               :: "v"(ldsb + 8192u),   "v"(pt.a[2]) : "memory");
  asm volatile("global_load_async_to_lds_b128 %0, %1, off"
               :: "v"(ldsb + 12288u),  "v"(pt.a[3]) : "memory");
}

__global__ __launch_bounds__(256)
void roialign_kernel(const float* __restrict__ f0, const float* __restrict__ f1,
                     const float* __restrict__ f2, const float* __restrict__ f3,
                     const float* __restrict__ boxes, float* __restrict__ out) {
  // sbuf[buffer][neighbor-row][tid] : 2 * 4 * 256 * 16B = 32KB
  __shared__ v4f sbuf[2][4][256];

  const int roi = blockIdx.x;        // 0..1023 (b*512+n)
  const int b   = roi >> 9;
  const int t   = threadIdx.x;       // 0..255
  const int c4  = t & 63;            // channel chunk (4 floats)
  const int grp = t >> 6;            // grid-point group 0..3 (wave-aligned)

  const float bx1 = boxes[roi * 4 + 0];
  const float by1 = boxes[roi * 4 + 1];
  const float bx2 = boxes[roi * 4 + 2];
  const float by2 = boxes[roi * 4 + 3];

  // FPN level (matches reference _roi_level): scale = 56/1024 = 0.0546875
  const float area = (by2 - by1) * (bx2 - bx1);
  const float q    = sqrtf(area) / (0.0546875f + 1e-7f);
  int lvl = (int)floorf(logf(q) / logf(2.0f));
  lvl = min(max(lvl, 0), NLVL - 1);

  const float* fp = f0; int S = 256;
  if (lvl == 1) { fp = f1; S = 128; }
  if (lvl == 2) { fp = f2; S = 64;  }
  if (lvl == 3) { fp = f3; S = 32;  }

  const float sm1   = (float)(S - 1);
  const float ybase = by1 * sm1;
  const float xbase = bx1 * sm1;
  const float dy    = (by2 - by1) * sm1 / 6.0f;   // (pool-1) == 6
  const float dx    = (bx2 - bx1) * sm1 / 6.0f;

  const unsigned long long fbase = (unsigned long long)(uintptr_t)fp;
  // LDS byte offset of this thread's chunk in buffer 0 (low 32 bits of the
  // flat shared address are the wave-relative LDS byte address on CDNA5).
  const unsigned lds0  = (unsigned)(uintptr_t)(&sbuf[0][0][0]);
  const unsigned ldsb0 = lds0 + (unsigned)t * 16u;

  v4f* __restrict__ outv = (v4f*)out;

  // Software pipeline: prefetch grid point p into buffer s while blending p-4.
  Pt cur = make_pt(fbase, S, b, ybase, dy, xbase, dx, grp, c4);
  issue4(cur, ldsb0);                 // buffer 0
  int s = 0;

  for (int p = grp; p < NPTS; p += 4) {
    Pt nxt = cur;
    const bool more = (p + 4) < NPTS;     // uniform per wave
    if (more) {
      nxt = make_pt(fbase, S, b, ybase, dy, xbase, dx, p + 4, c4);
      issue4(nxt, ldsb0 + (unsigned)((s ^ 1) * 16384));
      asm volatile("s_wait_asynccnt 0x4" ::: "memory");  // current 4 complete
    } else {
      asm volatile("s_wait_asynccnt 0x0" ::: "memory");
    }

    const v4f g00 = sbuf[s][0][t];
    const v4f g01 = sbuf[s][1][t];
    const v4f g10 = sbuf[s][2][t];
    const v4f g11 = sbuf[s][3][t];
    v4f o = g00 * cur.w00 + g01 * cur.w01 + g10 * cur.w10 + g11 * cur.w11;
    if (!cur.valid) o = (v4f){0.f, 0.f, 0.f, 0.f};

    // out[b,n,i,j,c]: flat v4f index = (roi*49 + p)*64 + c4; NT store keeps
    // the 51MB output stream from evicting the (L2-resident) feature maps.
    const size_t oidx = ((size_t)roi * NPTS + (size_t)p) * 64 + (size_t)c4;
    __builtin_nontemporal_store(o, outv + oidx);

    cur = nxt;
    s ^= 1;
  }
}

extern "C" void kernel_launch(void* const* d_in, const int* in_sizes, int n_in,
                              void* d_out, int out_size, void* d_ws, size_t ws_size,
                              hipStream_t stream) {
  const float* f0    = (const float*)d_in[0];
  const float* f1    = (const float*)d_in[1];
  const float* f2    = (const float*)d_in[2];
  const float* f3    = (const float*)d_in[3];
  const float* boxes = (const float*)d_in[4];
  float* out = (float*)d_out;

  const int nroi = in_sizes[4] / 4;   // B*N = 1024
  roialign_kernel<<<dim3(nroi), dim3(256), 0, stream>>>(f0, f1, f2, f3, boxes, out);
}